// AFNOBlock_19215683682437
// MI455X (gfx1250) — compile-verified
//
#include <hip/hip_runtime.h>
#include <hip/hip_bf16.h>
#include <math.h>

// ---------------- CDNA5 fp32 WMMA (16x16x4) ----------------
typedef __attribute__((ext_vector_type(2))) float v2f;
typedef __attribute__((ext_vector_type(8))) float v8f;

static __device__ __forceinline__ v8f wmma_f32_4(v2f a, v2f b, v8f c) {
  // D = A(16x4) * B(4x16) + C(16x16), fp32, wave32
  return __builtin_amdgcn_wmma_f32_16x16x4_f32(false, a, false, b, (short)0, c, false, false);
}

#define PI_F 3.14159265358979323846f

// ---------------- twiddle tables ----------------
// Fw[wf*64+w]  = exp(-2pi i w wf/64) / 64        (forward W-rfft, full ortho scale here)
// Fh[k*64+h]   = exp(-2pi i h k/64)              (forward H-DFT)
// Gh[h*64+k]   = exp(+2pi i h k/64) / 64         (inverse H-DFT, full inverse ortho scale here)
// CS[w*33+wf]  = {cw, sw} hermitian-fold weights: xs[w] = sum_wf Yr*cw - Yi*sw
__global__ void k_twiddle(float2* __restrict__ Fw, float2* __restrict__ Fh,
                          float2* __restrict__ Gh, float2* __restrict__ CS) {
  int t = threadIdx.x;
  const float s = 1.0f / 64.0f;
  for (int i = t; i < 33 * 64; i += 256) {
    int wf = i >> 6, w = i & 63;
    int ph = (w * wf) & 63;
    float a = (float)ph * (PI_F / 32.0f);
    Fw[i] = make_float2(cosf(a) * s, -sinf(a) * s);
  }
  for (int i = t; i < 64 * 64; i += 256) {
    int k = i >> 6, h = i & 63;
    int ph = (k * h) & 63;
    float a = (float)ph * (PI_F / 32.0f);
    Fh[i] = make_float2(cosf(a), -sinf(a));
    Gh[i] = make_float2(cosf(a) * s, sinf(a) * s);
  }
  for (int i = t; i < 64 * 33; i += 256) {
    int w = i / 33, wf = i % 33;
    float cw, sw;
    if (wf == 0)       { cw = 1.0f; sw = 0.0f; }
    else if (wf == 32) { cw = (w & 1) ? -1.0f : 1.0f; sw = 0.0f; }
    else {
      int ph = (w * wf) & 63;
      float a = (float)ph * (PI_F / 32.0f);
      cw = 2.0f * cosf(a); sw = 2.0f * sinf(a);
    }
    CS[i] = make_float2(cw, sw);
  }
}

// ---------------- layernorm over C=384, one wave per token ----------------
__global__ void k_ln(const float* __restrict__ in, const float* __restrict__ g,
                     const float* __restrict__ bta, float* __restrict__ out) {
  int wave = threadIdx.x >> 5, lane = threadIdx.x & 31;
  size_t token = (size_t)blockIdx.x * 8 + wave;
  const float* p = in + token * 384;
  float v[12]; float s = 0.f, s2 = 0.f;
#pragma unroll
  for (int j = 0; j < 12; j++) { v[j] = p[lane + j * 32]; s += v[j]; s2 += v[j] * v[j]; }
  for (int off = 16; off > 0; off >>= 1) { s += __shfl_xor(s, off); s2 += __shfl_xor(s2, off); }
  float mu = s * (1.f / 384.f);
  float var = s2 * (1.f / 384.f) - mu * mu;
  float rs = rsqrtf(var + 1e-5f);
  float* o = out + token * 384;
#pragma unroll
  for (int j = 0; j < 12; j++) { int c = lane + j * 32; o[c] = (v[j] - mu) * rs * g[c] + bta[c]; }
}

// ---------------- forward rfft along W: Xw[b,h,wf,c] ----------------
__global__ void k_fftw(const float* __restrict__ xn, const float2* __restrict__ Fw,
                       float* __restrict__ XwR, float* __restrict__ XwI) {
  __shared__ float sx[64 * 128];
  __shared__ float2 stw[33 * 64];
  int t = threadIdx.x;          // 128
  int bh = blockIdx.x;          // b*64+h
  int c0 = blockIdx.y * 128;
  for (int idx = t; idx < 64 * 128; idx += 128) {
    int w = idx >> 7, cc = idx & 127;
    sx[idx] = xn[((size_t)bh * 64 + w) * 384 + c0 + cc];
  }
  for (int idx = t; idx < 33 * 64; idx += 128) stw[idx] = Fw[idx];
  __syncthreads();
  int c = t;
  for (int wf = 0; wf < 33; wf++) {
    float ar = 0.f, ai = 0.f;
#pragma unroll 8
    for (int w = 0; w < 64; w++) {
      float v = sx[(w << 7) + c];
      float2 tw = stw[wf * 64 + w];
      ar = fmaf(v, tw.x, ar);
      ai = fmaf(v, tw.y, ai);
    }
    size_t o = ((size_t)bh * 33 + wf) * 384 + c0 + c;
    XwR[o] = ar; XwI[o] = ai;
  }
}

// ---------------- complex DFT along H (fwd with Fh, inv with Gh) ----------------
__global__ void k_dfth(const float* __restrict__ inR, const float* __restrict__ inI,
                       const float2* __restrict__ tab,
                       float* __restrict__ outR, float* __restrict__ outI) {
  __shared__ float sR[64 * 64], sI[64 * 64];
  int t = threadIdx.x;                 // 256
  int b = blockIdx.x / 33, wf = blockIdx.x % 33;
  int c0 = blockIdx.y * 64;
  for (int idx = t; idx < 64 * 64; idx += 256) {
    int h = idx >> 6, cc = idx & 63;
    size_t g = (((size_t)b * 64 + h) * 33 + wf) * 384 + c0 + cc;
    sR[idx] = inR[g]; sI[idx] = inI[g];
  }
  __syncthreads();
  int c = t & 63, kq = t >> 6;
  for (int kk = 0; kk < 16; kk++) {
    int k = kq * 16 + kk;
    float ar = 0.f, ai = 0.f;
    const float2* tp = tab + k * 64;
#pragma unroll 8
    for (int h = 0; h < 64; h++) {
      float xr = sR[(h << 6) + c], xi = sI[(h << 6) + c];
      float2 tw = tp[h];
      ar += xr * tw.x - xi * tw.y;
      ai += xr * tw.y + xi * tw.x;
    }
    size_t o = (((size_t)b * 64 + k) * 33 + wf) * 384 + c0 + c;
    outR[o] = ar; outI[o] = ai;
  }
}

// ---------------- spectral block matmul via WMMA ----------------
// Per (h,wf,block n): [Xr Xi](16x96, rows 8..15 zero) @ [[Wr Wi],[-Wi Wr]](96x96)
// -> [Or Oi], + complex bias, ReLU(re), ReLU(im).  6 waves, each owns 16 cols.
__global__ void k_spectral(const float* __restrict__ XfR, const float* __restrict__ XfI,
                           const float* __restrict__ w1, const float* __restrict__ b1,
                           float* __restrict__ OR, float* __restrict__ OI) {
  __shared__ float sA[16 * 97];
  int t = threadIdx.x;        // 192
  int pos = blockIdx.x;       // h*33+wf
  int h = pos / 33, wf = pos % 33;
  int n = blockIdx.y;
  for (int idx = t; idx < 16 * 96; idx += 192) {
    int row = idx / 96, col = idx % 96;
    float v = 0.f;
    if (row < 8) {
      size_t base = (((size_t)row * 64 + h) * 33 + wf) * 384 + n * 48;
      v = (col < 48) ? XfR[base + col] : XfI[base + col - 48];
    }
    sA[row * 97 + col] = v;
  }
  __syncthreads();
  int lane = t & 31, wv = t >> 5;
  int half = lane >> 4, r = lane & 15;
  int col = wv * 16 + r;
  const float* wb = w1 + ((size_t)pos * 8 + n) * 48 * 48 * 2;
  const float* p1; const float* p2; float s2;
  if (col < 48) { p1 = wb + col * 2;            p2 = wb + col * 2 + 1;        s2 = -1.f; }
  else          { p1 = wb + (col - 48) * 2 + 1; p2 = wb + (col - 48) * 2;     s2 =  1.f; }
  v8f acc = {};
#pragma unroll
  for (int kk = 0; kk < 12; kk++) {       // i in [0,48): [Wr | Wi]
    int i0 = kk * 4 + half * 2;
    v2f a, bv;
    a.x = sA[r * 97 + i0];      a.y = sA[r * 97 + i0 + 1];
    bv.x = p1[i0 * 96];         bv.y = p1[(i0 + 1) * 96];
    acc = wmma_f32_4(a, bv, acc);
  }
#pragma unroll
  for (int kk = 0; kk < 12; kk++) {       // i in [48,96): [-Wi | Wr]
    int i0 = kk * 4 + half * 2;
    v2f a, bv;
    a.x = sA[r * 97 + 48 + i0]; a.y = sA[r * 97 + 48 + i0 + 1];
    bv.x = s2 * p2[i0 * 96];    bv.y = s2 * p2[(i0 + 1) * 96];
    acc = wmma_f32_4(a, bv, acc);
  }
  if (half == 0) {  // rows 0..7 = batch; rows 8..15 are padding
    const float* bp = b1 + ((size_t)pos * 8 + n) * 48 * 2;
    float bb = (col < 48) ? bp[col * 2] : bp[(col - 48) * 2 + 1];
#pragma unroll
    for (int j = 0; j < 8; j++) {
      float v = fmaxf(acc[j] + bb, 0.f);
      size_t o = (((size_t)j * 64 + h) * 33 + wf) * 384 + n * 48;
      if (col < 48) OR[o + col] = v;
      else          OI[o + col - 48] = v;
    }
  }
}

// ---------------- inverse rfft along W (hermitian fold) ----------------
__global__ void k_ifftw(const float* __restrict__ YR, const float* __restrict__ YI,
                        const float2* __restrict__ CS, float* __restrict__ xs) {
  __shared__ float sR[33 * 128], sI[33 * 128];
  __shared__ float2 scs[64 * 33];
  int t = threadIdx.x;       // 256
  int bh = blockIdx.x;
  int c0 = blockIdx.y * 128;
  for (int idx = t; idx < 33 * 128; idx += 256) {
    int wf = idx >> 7, cc = idx & 127;
    size_t g = ((size_t)bh * 33 + wf) * 384 + c0 + cc;
    sR[idx] = YR[g]; sI[idx] = YI[g];
  }
  for (int idx = t; idx < 64 * 33; idx += 256) scs[idx] = CS[idx];
  __syncthreads();
  int c = t & 127, wq = t >> 7;
  for (int ww = 0; ww < 32; ww++) {
    int w = wq * 32 + ww;
    float acc = 0.f;
#pragma unroll
    for (int wf = 0; wf < 33; wf++) {
      float2 cs = scs[w * 33 + wf];
      acc += sR[(wf << 7) + c] * cs.x - sI[(wf << 7) + c] * cs.y;
    }
    xs[((size_t)bh * 64 + w) * 384 + c0 + c] = acc;
  }
}

// ---------------- tiled fp32 WMMA GEMM, 64x64 tile, KT=32, 8 waves ----------------
// MODE 0: out = GELU(A@B + bias);  MODE 1: out = A@B + bias + add0 + add1
template <int MODE>
__global__ void k_gemm(const float* __restrict__ A, const float* __restrict__ Bm,
                       const float* __restrict__ bias,
                       const float* __restrict__ add0, const float* __restrict__ add1,
                       float* __restrict__ out, int N, int K) {
  __shared__ float As[64 * 33];   // padded stride 33 (bank-conflict free frag reads)
  __shared__ float Bs[32 * 64];
  int t = threadIdx.x;            // 256
  int m0 = blockIdx.x * 64, n0 = blockIdx.y * 64;
  int lane = t & 31, wv = t >> 5;
  int half = lane >> 4, r = lane & 15;
  int mb = (wv & 3) * 16;         // wave's 16-row strip
  int nb = (wv >> 2) * 32;        // wave's 32-col strip (2 n-tiles)
  v8f acc0 = {}, acc1 = {};
  for (int k0 = 0; k0 < K; k0 += 32) {
    {
      int row = t >> 5, kk = t & 31;
#pragma unroll
      for (int i = 0; i < 8; i++) {
        int rr = row + i * 8;
        As[rr * 33 + kk] = A[((size_t)(m0 + rr)) * K + k0 + kk];
      }
      int br = t >> 6, bn = t & 63;
#pragma unroll
      for (int i = 0; i < 8; i++) {
        int rr = br + i * 4;
        Bs[rr * 64 + bn] = Bm[((size_t)(k0 + rr)) * N + n0 + bn];
      }
    }
    __syncthreads();
#pragma unroll
    for (int kk = 0; kk < 32; kk += 4) {
      int ki = kk + half * 2;
      v2f a, b0, b1v;
      a.x = As[(mb + r) * 33 + ki];        a.y = As[(mb + r) * 33 + ki + 1];
      b0.x = Bs[ki * 64 + nb + r];         b0.y = Bs[(ki + 1) * 64 + nb + r];
      b1v.x = Bs[ki * 64 + nb + 16 + r];   b1v.y = Bs[(ki + 1) * 64 + nb + 16 + r];
      acc0 = wmma_f32_4(a, b0, acc0);
      acc1 = wmma_f32_4(a, b1v, acc1);
    }
    __syncthreads();
  }
#pragma unroll
  for (int nt = 0; nt < 2; nt++) {
    v8f acc = nt ? acc1 : acc0;
    int col = n0 + nb + nt * 16 + r;
    float bcol = bias[col];
#pragma unroll
    for (int j = 0; j < 8; j++) {
      int m = m0 + mb + j + half * 8;
      size_t o = (size_t)m * N + col;
      float v = acc[j] + bcol;
      if (MODE == 0) {
        out[o] = 0.5f * v * (1.f + erff(v * 0.70710678118654752440f));
      } else {
        out[o] = v + add0[o] + add1[o];
      }
    }
  }
}

// ---------------- launch ----------------
extern "C" void kernel_launch(void* const* d_in, const int* in_sizes, int n_in,
                              void* d_out, int out_size, void* d_ws, size_t ws_size,
                              hipStream_t stream) {
  const float* x   = (const float*)d_in[0];
  const float* w1  = (const float*)d_in[1];
  const float* b1  = (const float*)d_in[2];
  const float* n1w = (const float*)d_in[3];
  const float* n1b = (const float*)d_in[4];
  const float* n2w = (const float*)d_in[5];
  const float* n2b = (const float*)d_in[6];
  const float* mw1 = (const float*)d_in[7];
  const float* mb1 = (const float*)d_in[8];
  const float* mw2 = (const float*)d_in[9];
  const float* mb2 = (const float*)d_in[10];
  float* out = (float*)d_out;
  float* ws = (float*)d_ws;

  // workspace layout (floats)
  float2* Fw = (float2*)(ws + 0);        // 33*64 complex
  float2* Fh = (float2*)(ws + 4224);     // 64*64
  float2* Gh = (float2*)(ws + 12416);    // 64*64
  float2* CS = (float2*)(ws + 20608);    // 64*33
  float* xn  = ws + 32768;               // 8*64*64*384   (reused as xm after irfft)
  float* XwR = xn  + 12582912;           // 8*64*33*384   (reused as O1R)
  float* XwI = XwR + 6488064;            //               (reused as O1I)
  float* XfR = XwI + 6488064;            //               (reused as Y1R)
  float* XfI = XfR + 6488064;            //               (reused as Y1I)
  float* xs  = XfI + 6488064;            // 8*64*64*384
  float* hdn = xs  + 12582912;           // 32768*1536

  k_twiddle<<<1, 256, 0, stream>>>(Fw, Fh, Gh, CS);
  k_ln<<<4096, 256, 0, stream>>>(x, n1w, n1b, xn);
  k_fftw<<<dim3(512, 3), 128, 0, stream>>>(xn, Fw, XwR, XwI);
  k_dfth<<<dim3(264, 6), 256, 0, stream>>>(XwR, XwI, Fh, XfR, XfI);
  k_spectral<<<dim3(2112, 8), 192, 0, stream>>>(XfR, XfI, w1, b1, XwR, XwI);
  k_dfth<<<dim3(264, 6), 256, 0, stream>>>(XwR, XwI, Gh, XfR, XfI);
  k_ifftw<<<dim3(512, 3), 256, 0, stream>>>(XfR, XfI, CS, xs);
  k_ln<<<4096, 256, 0, stream>>>(xs, n2w, n2b, xn);
  k_gemm<0><<<dim3(512, 24), 256, 0, stream>>>(xn, mw1, mb1, nullptr, nullptr, hdn, 1536, 384);
  k_gemm<1><<<dim3(512, 6), 256, 0, stream>>>(hdn, mw2, mb2, xs, x, out, 384, 1536);
}